// BondAngleGuidance_11562051961094
// MI455X (gfx1250) — compile-verified
//
#include <hip/hip_runtime.h>
#include <hip/hip_bf16.h>

typedef float v2f __attribute__((ext_vector_type(2)));
typedef float v8f __attribute__((ext_vector_type(8)));

#define D_MAX 16
#define WAVES_PER_BLOCK 8
#define RAD2DEG 57.29577951308232f

// ---------------- kernel 0: zero per-node counters ----------------
__global__ __launch_bounds__(256) void bag_init(int* __restrict__ cnt, int n) {
  int i = blockIdx.x * 256 + threadIdx.x;
  if (i < n) cnt[i] = 0;
}

// ---------------- kernel 1: build padded neighbor table from half-edges ----
__global__ __launch_bounds__(256) void bag_scatter(const int* __restrict__ etype,
                                                   const int* __restrict__ eidx,
                                                   int* __restrict__ nbr,
                                                   int* __restrict__ cnt, int E) {
  int e = blockIdx.x * 256 + threadIdx.x;
  if (e >= E) return;
  if (etype[e] == 0) return;
  int s = eidx[e];
  int d = eidx[E + e];
  int p = atomicAdd(&cnt[s], 1);
  if (p < D_MAX) nbr[s * D_MAX + p] = d;
  int q = atomicAdd(&cnt[d], 1);
  if (q < D_MAX) nbr[d * D_MAX + q] = s;
}

// ---------------- kernel 2: one wave32 per node, WMMA pairwise dots --------
__global__ __launch_bounds__(256) void bag_angles(const float* __restrict__ x,
                                                  const int* __restrict__ nbr,
                                                  const int* __restrict__ cnt,
                                                  float* __restrict__ partials,
                                                  int n) {
  __shared__ int   s_sort[WAVES_PER_BLOCK][D_MAX];
  __shared__ float s_wsum[WAVES_PER_BLOCK];

  const int lane = threadIdx.x & 31;
  const int wave = threadIdx.x >> 5;
  const int node_raw = blockIdx.x * WAVES_PER_BLOCK + wave;
  const bool live = node_raw < n;
  const int node = live ? node_raw : (n - 1);
  const int j = lane & 15;          // neighbor slot handled by this lane (column N)

  int c = cnt[node];
  c = (c > D_MAX) ? D_MAX : c;
  const bool jvalid = j < c;
  int nid0 = jvalid ? nbr[node * D_MAX + j] : node;  // self => v = 0

  // Canonical deterministic slot ordering: rank-sort the 16 ids within the wave.
  // Invalid slots get key >= n<<5 so they sort to the tail (slots c..15).
  int key = (((jvalid ? nid0 : n) << 5) | j);
  int rank = 0;
#pragma unroll
  for (int k = 0; k < 16; ++k) {
    int ok = __shfl(key, k, 32);
    rank += (ok < key) ? 1 : 0;
  }
  if (lane < 16) s_sort[wave][rank] = nid0;
  __syncthreads();
  const int nid = s_sort[wave][j];

  const float cx = x[3 * node + 0], cy = x[3 * node + 1], cz = x[3 * node + 2];
  const float vx = x[3 * nid + 0] - cx;
  const float vy = x[3 * nid + 1] - cy;
  const float vz = x[3 * nid + 2] - cz;
  const float norm2 = vx * vx + vy * vy + vz * vz;

  // A (16x4, f32): lanes 0-15 hold row M=j comps K=0,1 ; lanes 16-31 comps K=2,3.
  // B (4x16) is the transpose-symmetric layout, so the same per-lane data serves
  // both operands for dots = V * V^T.
  v2f a;
  a.x = (lane < 16) ? vx : vz;
  a.y = (lane < 16) ? vy : 0.0f;
  v8f acc = {};
  v8f dots = __builtin_amdgcn_wmma_f32_16x16x4_f32(
      /*neg_a=*/false, a, /*neg_b=*/false, a,
      /*c_mod=*/(short)0, acc, /*reuse_a=*/false, /*reuse_b=*/false);

  // D layout: VGPR r, lanes 0-15 -> (M=r,   N=lane),
  //                   lanes 16-31-> (M=r+8, N=lane-16)
  // Use hardware v_rsq_f32 instead of sqrt+IEEE-divide: cos = dot*rsq(m)*rsq(n).
  // Invalid slots (norm2 == 0) get inv = 0 -> cos = 0, matching the masked path.
  const float inv_n = (norm2 > 0.0f) ? __builtin_amdgcn_rsqf(norm2) : 0.0f;
  const int mbase = (lane < 16) ? 0 : 8;
  float sum = 0.0f;
#pragma unroll
  for (int r = 0; r < 8; ++r) {
    int m = mbase + r;
    float inv_m = __shfl(inv_n, m, 32);    // slot m's 1/|v| lives in lane m
    float cosv = dots[r] * inv_m * inv_n;
    cosv = fminf(1.0f, fmaxf(-1.0f, cosv));
    float ang = acosf(cosv) * RAD2DEG;
    float drift = 0.1f * fmaxf(100.0f - ang, 0.0f);
    bool take = (m < j) && (j < c);        // row<col once per unordered pair; m<j<c => m valid
    sum += take ? drift : 0.0f;
  }
  if (!live) sum = 0.0f;

  // wave32 butterfly reduction
#pragma unroll
  for (int off = 16; off >= 1; off >>= 1)
    sum += __shfl_xor(sum, off, 32);
  if (lane == 0) s_wsum[wave] = sum;
  __syncthreads();
  if (threadIdx.x == 0) {
    float t = 0.0f;
    for (int w = 0; w < WAVES_PER_BLOCK; ++w) t += s_wsum[w];
    partials[blockIdx.x] = t;
  }
}

// ---------------- kernel 3: deterministic final reduction ------------------
__global__ __launch_bounds__(256) void bag_reduce(const float* __restrict__ partials,
                                                  int n, float* __restrict__ out) {
  __shared__ float sh[256];
  float s = 0.0f;
  for (int i = threadIdx.x; i < n; i += 256) s += partials[i];
  sh[threadIdx.x] = s;
  __syncthreads();
  for (int st = 128; st > 0; st >>= 1) {
    if (threadIdx.x < st) sh[threadIdx.x] += sh[threadIdx.x + st];
    __syncthreads();
  }
  if (threadIdx.x == 0) out[0] = sh[0];
}

extern "C" void kernel_launch(void* const* d_in, const int* in_sizes, int n_in,
                              void* d_out, int out_size, void* d_ws, size_t ws_size,
                              hipStream_t stream) {
  (void)n_in; (void)out_size; (void)ws_size;
  const float* x     = (const float*)d_in[0];  // [N,3] f32
  const int*   etype = (const int*)d_in[1];    // [E]   i32
  const int*   eidx  = (const int*)d_in[2];    // [2,E] i32 (src row, then dst row)

  const int N = in_sizes[0] / 3;
  const int E = in_sizes[1];

  // workspace layout: cnt[N] | nbr[N*16] | partials[numBlocks]
  int*   cnt      = (int*)d_ws;
  int*   nbr      = cnt + N;
  const int nblocks = (N + WAVES_PER_BLOCK - 1) / WAVES_PER_BLOCK;
  float* partials = (float*)(nbr + (size_t)N * D_MAX);

  bag_init<<<(N + 255) / 256, 256, 0, stream>>>(cnt, N);
  bag_scatter<<<(E + 255) / 256, 256, 0, stream>>>(etype, eidx, nbr, cnt, E);
  bag_angles<<<nblocks, 256, 0, stream>>>(x, nbr, cnt, partials, N);
  bag_reduce<<<1, 256, 0, stream>>>(partials, nblocks, (float*)d_out);
}